// RiemannianGraph_HRSNN_71365176590554
// MI455X (gfx1250) — compile-verified
//
#include <hip/hip_runtime.h>

typedef __attribute__((ext_vector_type(2))) float v2f;
typedef __attribute__((ext_vector_type(8))) float v8f;

// ---------------------------------------------------------------------------
// FP32 WMMA 16x16x4:  D = A(16x4) * B(4x16) + C(16x16), all f32.
// ---------------------------------------------------------------------------
__device__ __forceinline__ v8f wmma4(v2f a, v2f b, v8f c) {
  return __builtin_amdgcn_wmma_f32_16x16x4_f32(
      /*neg_a=*/false, a, /*neg_b=*/false, b,
      /*c_mod=*/(short)0, c, /*reuse_a=*/false, /*reuse_b=*/false);
}

// A fragment (16x4 f32): lanes 0-15 rows M, VGPR j holds K = 2*(lane>=16)+j.
__device__ __forceinline__ v2f frag_a(const float* A, int stride, int r0, int k0, int lane) {
  int half = lane >> 4, m = lane & 15;
  const float* p = A + (r0 + m) * stride + k0 + 2 * half;
  v2f a; a[0] = p[0]; a[1] = p[1];
  return a;
}
// B fragment (4x16 f32): lanes 0-15 cols N, VGPR j holds K = 2*(lane>=16)+j.
__device__ __forceinline__ v2f frag_b(const float* B, int stride, int k0, int c0, int lane) {
  int half = lane >> 4, n = lane & 15;
  v2f b;
  b[0] = B[(k0 + 2 * half + 0) * stride + c0 + n];
  b[1] = B[(k0 + 2 * half + 1) * stride + c0 + n];
  return b;
}
// C/D fragment (16x16 f32): VGPR v, lane group -> element (M = v + 8*(lane>=16), N = lane&15).
__device__ __forceinline__ void store_c(float* C, int stride, int r0, int c0, int lane, v8f c) {
  int half = lane >> 4, n = lane & 15;
#pragma unroll
  for (int v = 0; v < 8; ++v)
    C[(r0 + v + 8 * half) * stride + c0 + n] = c[v];
}

// Triple LIF (snnTorch Leaky, reset='subtract', detached reset), shared betas/thrs.
__device__ __forceinline__ float lif3(float c, float& ma, float& mn, float& mm) {
  float s = 0.0f;
  float r0 = (ma > 0.8f) ? 1.0f : 0.0f;
  ma = 0.80f * ma + c - r0 * 0.8f;
  s += (ma > 0.8f) ? 1.0f : 0.0f;
  float r1 = (mn > 1.0f) ? 1.0f : 0.0f;
  mn = 0.90f * mn + c - r1 * 1.0f;
  s += (mn > 1.0f) ? 1.0f : 0.0f;
  float r2 = (mm > 1.5f) ? 1.0f : 0.0f;
  mm = 0.95f * mm + c - r2 * 1.5f;
  s += (mm > 1.5f) ? 1.0f : 0.0f;
  return s;
}

// ---------------------------------------------------------------------------
// Kernel 1: full 15-step recurrence for one batch element per block.
// 256 threads = 8 wave32. All membrane state AND all weight B-fragments live
// in registers across the time loop; after the prologue the loop does zero
// global memory traffic (LDS fragments + WMMA only).
// Writes dp_feat[B,4096] = (mean last5 - mean first5) into workspace.
// ---------------------------------------------------------------------------
__global__ __launch_bounds__(256) void hrsnn_recurrent(
    const float* __restrict__ Lg,  const float* __restrict__ Xg,
    const float* __restrict__ Wp,  const float* __restrict__ bp,
    const float* __restrict__ Wg1, const float* __restrict__ bg1,
    const float* __restrict__ Ws1, const float* __restrict__ bs1,
    const float* __restrict__ Wg2, const float* __restrict__ bg2,
    const float* __restrict__ Ws2, const float* __restrict__ bs2,
    float* __restrict__ dp_out)
{
  __shared__ float sL[64 * 64];     // Laplacian, resident across all steps
  __shared__ float sX[15 * 64];     // input sequence for this batch element
  __shared__ float sXP[64 * 16];    // projected input
  __shared__ float sG1[64 * 32];    // xp @ W_gcn1
  __shared__ float sSpk1[64 * 32];  // layer-1 output
  __shared__ float sG2[64 * 64];    // spk1 @ W_gcn2

  const int b = blockIdx.x;
  const int tid = threadIdx.x;
  const int lane = tid & 31;
  const int w = tid >> 5;

  for (int i = tid; i < 64 * 64; i += 256) sL[i] = Lg[(size_t)b * 4096 + i];
  for (int i = tid; i < 15 * 64; i += 256) sX[i] = Xg[(size_t)b * (15 * 64) + i];

  // ---- loop-invariant register preload -------------------------------------
  // Projection row for this thread's fixed feature column (i & 15 == tid & 15
  // for every i = tid + 256*r since 256 % 16 == 0).
  const float wpv = Wp[tid & 15];
  const float bpv = bp[tid & 15];

  // Layer-1 tile per wave (64x32 = 4x2 tiles); layer-2 tiles w and w+8 (4x4 tiles).
  const int r1 = (w >> 1) * 16, c1 = (w & 1) * 16;
  const int col1 = c1 + (lane & 15);
  const float bg1v = bg1[col1];
  const float bs1v = bs1[col1];

  v2f fWg1[4], fWs1[4];                 // [16,32] weights: K=16 -> 4 fragments
#pragma unroll
  for (int kk = 0; kk < 4; ++kk) {
    fWg1[kk] = frag_b(Wg1, 32, kk * 4, c1, lane);
    fWs1[kk] = frag_b(Ws1, 32, kk * 4, c1, lane);
  }

  int r2t[2], c2t[2];
  float bg2v[2], bs2v[2];
  v2f fWg2[2][8], fWs2[2][8];           // [32,64] weights: K=32 -> 8 fragments
#pragma unroll
  for (int it = 0; it < 2; ++it) {
    int tt = w + it * 8;
    r2t[it] = (tt >> 2) * 16;
    c2t[it] = (tt & 3) * 16;
    int col = c2t[it] + (lane & 15);
    bg2v[it] = bg2[col];
    bs2v[it] = bs2[col];
#pragma unroll
    for (int kk = 0; kk < 8; ++kk) {
      fWg2[it][kk] = frag_b(Wg2, 64, kk * 4, c2t[it], lane);
      fWs2[it][kk] = frag_b(Ws2, 64, kk * 4, c2t[it], lane);
    }
  }

  const v8f vzero = {};
  v8f m1a = vzero, m1n = vzero, m1m = vzero;            // layer-1 LIF membranes
  v8f m2a[2], m2n[2], m2m[2], ml[2], acc[2];            // layer-2 LIF + leaky + dp acc
#pragma unroll
  for (int it = 0; it < 2; ++it) {
    m2a[it] = vzero; m2n[it] = vzero; m2m[it] = vzero;
    ml[it] = vzero;  acc[it] = vzero;
  }
  __syncthreads();

  // ---- recurrent loop ------------------------------------------------------
  for (int t = 0; t < 15; ++t) {
    // xp = x_t * W_proj + b_proj   (outer product, [64,16])
    for (int i = tid; i < 64 * 16; i += 256)
      sXP[i] = sX[t * 64 + (i >> 4)] * wpv + bpv;
    __syncthreads();

    // G1 = xp @ W_gcn1   ([64,16] @ [16,32])
    {
      v8f g = vzero;
#pragma unroll
      for (int kk = 0; kk < 4; ++kk)
        g = wmma4(frag_a(sXP, 16, r1, kk * 4, lane), fWg1[kk], g);
      store_c(sG1, 32, r1, c1, lane, g);
    }
    __syncthreads();

    // cur1 = L @ G1 + b1 ; skip1 = xp @ Ws1 + bs1 ; triple LIF -> spk1
    {
      v8f cur = vzero;
#pragma unroll
      for (int k = 0; k < 64; k += 4)
        cur = wmma4(frag_a(sL, 64, r1, k, lane), frag_b(sG1, 32, k, c1, lane), cur);
      v8f sk = vzero;
#pragma unroll
      for (int kk = 0; kk < 4; ++kk)
        sk = wmma4(frag_a(sXP, 16, r1, kk * 4, lane), fWs1[kk], sk);
      v8f spk = vzero;
#pragma unroll
      for (int v = 0; v < 8; ++v) {
        float c = cur[v] + bg1v;
        float ma = m1a[v], mn = m1n[v], mm = m1m[v];
        float s = lif3(c, ma, mn, mm);
        m1a[v] = ma; m1n[v] = mn; m1m[v] = mm;
        spk[v] = s + sk[v] + bs1v;
      }
      store_c(sSpk1, 32, r1, c1, lane, spk);
    }
    __syncthreads();

    // G2 = spk1 @ W_gcn2   ([64,32] @ [32,64]) — 2 tiles per wave
#pragma unroll
    for (int it = 0; it < 2; ++it) {
      v8f g = vzero;
#pragma unroll
      for (int kk = 0; kk < 8; ++kk)
        g = wmma4(frag_a(sSpk1, 32, r2t[it], kk * 4, lane), fWg2[it][kk], g);
      store_c(sG2, 64, r2t[it], c2t[it], lane, g);
    }
    __syncthreads();

    // cur2 = L @ G2 + b2 ; skip2 = spk1 @ Ws2 + bs2 ; LIF ; mem_li ; dp accumulate
#pragma unroll
    for (int it = 0; it < 2; ++it) {
      v8f cur = vzero;
#pragma unroll
      for (int k = 0; k < 64; k += 4)
        cur = wmma4(frag_a(sL, 64, r2t[it], k, lane), frag_b(sG2, 64, k, c2t[it], lane), cur);
      v8f sk = vzero;
#pragma unroll
      for (int kk = 0; kk < 8; ++kk)
        sk = wmma4(frag_a(sSpk1, 32, r2t[it], kk * 4, lane), fWs2[it][kk], sk);
#pragma unroll
      for (int v = 0; v < 8; ++v) {
        float c = cur[v] + bg2v[it];
        float ma = m2a[it][v], mn = m2n[it][v], mm = m2m[it][v];
        float s = lif3(c, ma, mn, mm);
        m2a[it][v] = ma; m2n[it][v] = mn; m2m[it][v] = mm;
        float spk2 = s + sk[v] + bs2v[it];
        float m = 0.9f * ml[it][v] + spk2;
        ml[it][v] = m;
        if (t < 5)        acc[it][v] = acc[it][v] - m;
        else if (t >= 10) acc[it][v] = acc[it][v] + m;
      }
    }
    __syncthreads();
  }

  // dp_feat = (sum_last5 - sum_first5)/5 ; layout [B, node*64 + feat]
#pragma unroll
  for (int it = 0; it < 2; ++it) {
    int half = lane >> 4, n = lane & 15;
#pragma unroll
    for (int v = 0; v < 8; ++v) {
      int node = r2t[it] + v + 8 * half;
      int feat = c2t[it] + n;
      dp_out[(size_t)b * 4096 + node * 64 + feat] = acc[it][v] * 0.2f;
    }
  }
}

// ---------------------------------------------------------------------------
// Kernel 2: decoder. out = relu(dp @ W_c1 + b_c1) @ W_c2 + b_c2.
// One block = 16 batch rows; 8 waves each own a 16-wide column tile of the
// 128-wide hidden layer; K=4096 staged through LDS in chunks of 128.
// ---------------------------------------------------------------------------
__global__ __launch_bounds__(256) void hrsnn_decoder(
    const float* __restrict__ dp,  const float* __restrict__ Wc1,
    const float* __restrict__ bc1, const float* __restrict__ Wc2,
    const float* __restrict__ bc2, float* __restrict__ out)
{
  __shared__ float sA[16 * 128];
  __shared__ float sH[16 * 128];
  __shared__ float sW2[128 * 4];

  const int tid = threadIdx.x;
  const int lane = tid & 31;
  const int w = tid >> 5;
  const int row0 = blockIdx.x * 16;
  const int c0 = w * 16;

  const v8f vzero = {};
  v8f acc8 = vzero;

  for (int i = tid; i < 128 * 4; i += 256) sW2[i] = Wc2[i];

  for (int kc = 0; kc < 4096; kc += 128) {
    __syncthreads();
    for (int i = tid; i < 16 * 128; i += 256) {
      int r = i >> 7, k = i & 127;
      sA[i] = dp[(size_t)(row0 + r) * 4096 + kc + k];
    }
    __syncthreads();
#pragma unroll
    for (int k = 0; k < 128; k += 4)
      acc8 = wmma4(frag_a(sA, 128, 0, k, lane),
                   frag_b(Wc1 + (size_t)kc * 128, 128, k, c0, lane), acc8);
  }

  {
    int n = lane & 15;
    float bb = bc1[c0 + n];
    v8f h = vzero;
#pragma unroll
    for (int v = 0; v < 8; ++v) h[v] = fmaxf(acc8[v] + bb, 0.0f);
    store_c(sH, 128, 0, c0, lane, h);
  }
  __syncthreads();

  if (tid < 64) {
    int r = tid >> 2, c = tid & 3;
    float s = bc2[c];
    for (int k = 0; k < 128; ++k) s += sH[r * 128 + k] * sW2[k * 4 + c];
    out[(size_t)(row0 + r) * 4 + c] = s;
  }
}

// ---------------------------------------------------------------------------
extern "C" void kernel_launch(void* const* d_in, const int* in_sizes, int n_in,
                              void* d_out, int out_size, void* d_ws, size_t ws_size,
                              hipStream_t stream) {
  (void)n_in; (void)out_size; (void)ws_size;
  const float* L   = (const float*)d_in[0];
  const float* X   = (const float*)d_in[1];
  const float* Wp  = (const float*)d_in[2];
  const float* bp  = (const float*)d_in[3];
  const float* Wg1 = (const float*)d_in[4];
  const float* bg1 = (const float*)d_in[5];
  const float* Ws1 = (const float*)d_in[6];
  const float* bs1 = (const float*)d_in[7];
  const float* Wg2 = (const float*)d_in[8];
  const float* bg2 = (const float*)d_in[9];
  const float* Ws2 = (const float*)d_in[10];
  const float* bs2 = (const float*)d_in[11];
  const float* Wc1 = (const float*)d_in[12];
  const float* bc1 = (const float*)d_in[13];
  const float* Wc2 = (const float*)d_in[14];
  const float* bc2 = (const float*)d_in[15];

  const int B = in_sizes[0] / (64 * 64);     // 2048
  float* dp = (float*)d_ws;                  // needs B*4096 floats (32 MiB)

  hrsnn_recurrent<<<B, 256, 0, stream>>>(L, X, Wp, bp, Wg1, bg1, Ws1, bs1,
                                         Wg2, bg2, Ws2, bs2, dp);
  hrsnn_decoder<<<B / 16, 256, 0, stream>>>(dp, Wc1, bc1, Wc2, bc2, (float*)d_out);
}